// Scatches_75033078661777
// MI455X (gfx1250) — compile-verified
//
#include <hip/hip_runtime.h>
#include <hip/hip_bf16.h>
#include <stdint.h>

// Problem constants (match reference)
#define C_CH   3
#define COLS   3072
#define ROWS   3072
#define PLANE  (COLS * ROWS)            // 9,437,184 pixels
#define PLANEQ (PLANE / 4)              // 2,359,296 float4 groups per channel
#define NQ_TOTAL (C_CH * PLANEQ)        // 7,077,888 float4 chunks total
#define MASK_WORDS (PLANE / 32)         // 294,912 u32
#define MASK_BYTES (MASK_WORDS * 4)     // 1,179,648 B

// reduce1 geometry: 1536 blocks * 256 threads = 393,216 lanes; 7,077,888/393,216 = 18 exact
#define GRID_R 1536
#define ITERS_R 18
#define STRIDE_R (GRID_R * 256)

typedef float nfloat4 __attribute__((ext_vector_type(4)));   // native vec for NT stores

// ---- CDNA5 async global->LDS (ASYNCcnt path) -------------------------------
#define ASYNC_LOAD_B128(ldsoff, gaddr)                                         \
  asm volatile("global_load_async_to_lds_b128 %0, %1, off"                     \
               :: "v"(ldsoff), "v"(gaddr) : "memory")
#define WAIT_ASYNC(lit)                                                        \
  asm volatile("s_wait_asynccnt " #lit ::: "memory")

// ---------------------------------------------------------------------------
__global__ void clear_mask_kernel(uint4* __restrict__ mask16) {
  unsigned g = blockIdx.x * 256u + threadIdx.x;           // 73,728 uint4 words
  if (g < MASK_WORDS / 4) {
    uint4 z; z.x = 0u; z.y = 0u; z.z = 0u; z.w = 0u;
    mask16[g] = z;
  }
}

// One lane per crack; exact replica of the reference scan semantics:
// pixel painted at the PRE-update (x,y) while !done; endpoint never painted.
__global__ void paint_kernel(const int* __restrict__ xs_, const int* __restrict__ xe_,
                             const int* __restrict__ ys_, const int* __restrict__ ye_,
                             const int* __restrict__ len_, const int* __restrict__ wid_,
                             unsigned int* __restrict__ mask, int ncracks) {
  int t = threadIdx.x;
  if (t >= ncracks) return;
  int x = xs_[t], xe = xe_[t];
  int y = ys_[t], ye = ye_[t];
  int length = len_[t], width = wid_[t];
  if (length > 3) length = 3;   // reference dl loop caps at MAX_LENGTH-2 = 2
  if (width  > 3) width  = 3;   // reference dw loop caps at MAX_WIDTH-2  = 2

  int dx = (xe > x) ? (xe - x) : (x - xe);
  int dy = (ye > y) ? (ye - y) : (y - ye);
  int sx = (x < xe) ? 1 : -1;
  int sy = (y < ye) ? 1 : -1;
  int err = dx - dy;
  bool done = (x == xe) && (y == ye);

  for (int s = 0; s < (COLS + ROWS) && !done; ++s) {
    // paint rectangle rows yy = y..y+width-1, cols x..x+length-1 (clipped high side)
    int nb = length;
    if (nb > ROWS - x) nb = ROWS - x;   // x in [0,3071] during march -> nb >= 1
    unsigned long long run = (1ull << nb) - 1ull;
    #pragma unroll
    for (int dw = 0; dw < 3; ++dw) {
      int yy = y + dw;
      if (dw < width && yy < COLS) {
        unsigned bitpos = (unsigned)(yy * ROWS + x);
        unsigned long long m = run << (bitpos & 31u);
        unsigned w0 = bitpos >> 5;
        atomicOr(&mask[w0], (unsigned)m);
        unsigned hi = (unsigned)(m >> 32);
        if (hi) atomicOr(&mask[w0 + 1], hi);   // spill stays within same row (3072 % 32 == 0)
      }
    }
    // advance (Bresenham, both axes may move in one step — matches reference)
    int e2 = 2 * err;
    bool cx = e2 > -dy;
    bool cy = e2 <  dx;
    if (cx) { err -= dy; x += sx; }
    if (cy) { err += dx; y += sy; }
    done = (x == xe) && (y == ye);
  }
}

// Global max with 4-deep async global->LDS pipeline (ASYNCcnt), wave32 reduce.
// Regular-temporal path on purpose: this pass warms the 192 MB L2 with inp
// (113 MB fits), so the later apply pass reads it at L2 bandwidth.
__global__ void reduce1_kernel(const float4* __restrict__ in, float* __restrict__ partial) {
  __shared__ float4 stage[8][4][32];   // 8 waves x 4 buffers x 32 lanes x 16B = 16 KB
  __shared__ float wmax[8];

  const int lane = threadIdx.x & 31;
  const int w    = threadIdx.x >> 5;
  const unsigned tid = blockIdx.x * 256u + threadIdx.x;

  uint32_t ldsbase = (uint32_t)(uintptr_t)&stage[w][0][lane];   // low 32 bits = LDS offset
  const float4* base = in + tid;

  // prologue: 3 chunks in flight
  ASYNC_LOAD_B128(ldsbase + 0u * 512u, (unsigned long long)(uintptr_t)(base + 0 * STRIDE_R));
  ASYNC_LOAD_B128(ldsbase + 1u * 512u, (unsigned long long)(uintptr_t)(base + 1 * STRIDE_R));
  ASYNC_LOAD_B128(ldsbase + 2u * 512u, (unsigned long long)(uintptr_t)(base + 2 * STRIDE_R));

  float4 m4;
  m4.x = -__builtin_inff(); m4.y = -__builtin_inff();
  m4.z = -__builtin_inff(); m4.w = -__builtin_inff();

  #pragma unroll
  for (int k = 0; k < ITERS_R; ++k) {
    if (k <= ITERS_R - 4) {
      ASYNC_LOAD_B128(ldsbase + (unsigned)(((k + 3) & 3) * 512),
                      (unsigned long long)(uintptr_t)(base + (k + 3) * STRIDE_R));
      WAIT_ASYNC(0x3);
    } else if (k == ITERS_R - 3) { WAIT_ASYNC(0x2); }
    else if (k == ITERS_R - 2)   { WAIT_ASYNC(0x1); }
    else                         { WAIT_ASYNC(0x0); }

    float4 v = stage[w][k & 3][lane];   // ds_load_b128
    m4.x = fmaxf(m4.x, v.x); m4.y = fmaxf(m4.y, v.y);
    m4.z = fmaxf(m4.z, v.z); m4.w = fmaxf(m4.w, v.w);
  }

  float m = fmaxf(fmaxf(m4.x, m4.y), fmaxf(m4.z, m4.w));
  #pragma unroll
  for (int o = 16; o > 0; o >>= 1) m = fmaxf(m, __shfl_xor(m, o, 32));
  if (lane == 0) wmax[w] = m;
  __syncthreads();
  if (threadIdx.x == 0) {
    float r = wmax[0];
    #pragma unroll
    for (int i = 1; i < 8; ++i) r = fmaxf(r, wmax[i]);
    partial[blockIdx.x] = r;
  }
}

__global__ void reduce2_kernel(const float* __restrict__ partial, float* __restrict__ valp) {
  __shared__ float wm[8];
  float m = -__builtin_inff();
  for (int i = threadIdx.x; i < GRID_R; i += 256) m = fmaxf(m, partial[i]);
  #pragma unroll
  for (int o = 16; o > 0; o >>= 1) m = fmaxf(m, __shfl_xor(m, o, 32));
  if ((threadIdx.x & 31) == 0) wm[threadIdx.x >> 5] = m;
  __syncthreads();
  if (threadIdx.x == 0) {
    float r = wm[0];
    #pragma unroll
    for (int i = 1; i < 8; ++i) r = fmaxf(r, wm[i]);
    *valp = r;
  }
}

// Streaming where(mask, val, inp): 4 pixels/thread x 3 channels.
// Loads are regular-temporal (hit the L2 warmed by reduce1); stores are
// NON-temporal (th:TH_STORE_NT) so the 113 MB output stream does not evict
// the resident inp lines from the 192 MB L2.
__global__ void apply_kernel(const float4* __restrict__ in, nfloat4* __restrict__ out,
                             const unsigned int* __restrict__ mask,
                             const float* __restrict__ valp) {
  unsigned g = blockIdx.x * 256u + threadIdx.x;   // 0 .. PLANEQ-1 (exact grid)
  const float val = *valp;
  unsigned word = mask[g >> 3];
  unsigned bits = (word >> ((g & 7u) << 2)) & 0xFu;
  #pragma unroll
  for (int c = 0; c < C_CH; ++c) {
    float4 v = in[(unsigned)c * PLANEQ + g];
    nfloat4 r;
    r.x = (bits & 1u) ? val : v.x;
    r.y = (bits & 2u) ? val : v.y;
    r.z = (bits & 4u) ? val : v.z;
    r.w = (bits & 8u) ? val : v.w;
    __builtin_nontemporal_store(r, &out[(unsigned)c * PLANEQ + g]);
  }
}

extern "C" void kernel_launch(void* const* d_in, const int* in_sizes, int n_in,
                              void* d_out, int out_size, void* d_ws, size_t ws_size,
                              hipStream_t stream) {
  (void)n_in; (void)out_size; (void)ws_size;
  const float* inp = (const float*)d_in[0];
  const int* xs = (const int*)d_in[1];
  const int* xe = (const int*)d_in[2];
  const int* ys = (const int*)d_in[3];
  const int* ye = (const int*)d_in[4];
  const int* ln = (const int*)d_in[5];
  const int* wd = (const int*)d_in[6];
  float* out = (float*)d_out;
  const int ncracks = in_sizes[5];

  // workspace layout: [mask bitmap | per-block partials | val]
  unsigned int* mask = (unsigned int*)d_ws;
  float* partial = (float*)((char*)d_ws + MASK_BYTES);
  float* valp = partial + GRID_R;

  clear_mask_kernel<<<(MASK_WORDS / 4 + 255) / 256, 256, 0, stream>>>((uint4*)mask);
  paint_kernel<<<1, 64, 0, stream>>>(xs, xe, ys, ye, ln, wd, mask, ncracks);
  reduce1_kernel<<<GRID_R, 256, 0, stream>>>((const float4*)inp, partial);
  reduce2_kernel<<<1, 256, 0, stream>>>(partial, valp);
  apply_kernel<<<PLANEQ / 256, 256, 0, stream>>>((const float4*)inp, (nfloat4*)out, mask, valp);
}